// GetWeightMatrix_25812753449051
// MI455X (gfx1250) — compile-verified
//
#include <hip/hip_runtime.h>
#include <hip/hip_bf16.h>

#define CC 21
#define WW 512
#define HH 512
#define WH (WW * HH)
#define PENALTY 1.5f

#define TX 32           // threads along h
#define TY 8            // threads along w
#define PXT 4           // pixels per thread along h (b128 path)
#define TILE_W (TX * PXT + 4)   // 132 (128 + 2*halo)
#define TILE_H (TY + 4)         // 12  (8 + 2*halo)

typedef float v4f __attribute__((ext_vector_type(4)));

__global__ __launch_bounds__(TX * TY)
void GetWeightMatrix_25812753449051_kernel(const float* __restrict__ x,
                                           const int* __restrict__ y,
                                           float* __restrict__ out)
{
    __shared__ int tile[TILE_H * TILE_W];

    const int tx  = threadIdx.x;
    const int ty  = threadIdx.y;
    const int tid = ty * TX + tx;
    const int h0  = blockIdx.x * (TX * PXT);
    const int w0  = blockIdx.y * TY;
    const int b   = blockIdx.z;

    // ---------------------------------------------------------------
    // Stage y tile (+2 halo, zero-padded like nn.Unfold) into LDS via
    // the CDNA5 async-to-LDS path (tracked with ASYNCcnt). OOB halo
    // cells are zeroed with plain DS stores (disjoint addresses).
    // y keeps default TH=RT: halo lines are reused by up to 4 blocks,
    // and with the x stream marked NT they stay resident in L2.
    // ---------------------------------------------------------------
    const int* yb = y + (size_t)b * WH;   // uniform -> SGPR pair base
    for (int i = tid; i < TILE_H * TILE_W; i += TX * TY) {
        const int r  = i / TILE_W;
        const int c  = i - r * TILE_W;
        const int gw = w0 - 2 + r;
        const int gh = h0 - 2 + c;
        const unsigned lds = (unsigned)(size_t)&tile[i];  // low 32b = LDS offset
        if ((unsigned)gw < (unsigned)WW && (unsigned)gh < (unsigned)HH) {
            const unsigned goff = (((unsigned)gw << 9) + (unsigned)gh) << 2;
            asm volatile("global_load_async_to_lds_b32 %0, %1, %2"
                         :: "v"(lds), "v"(goff), "s"(yb)
                         : "memory");
        } else {
            tile[i] = 0;  // zero padding (label 0 is a valid class -> must be 0)
        }
    }

    // ---------------------------------------------------------------
    // Argmax over 21 channels for 4 consecutive h-pixels per thread.
    // x is [B, C, W, H]: per-channel stride WH, h-contiguous -> b128
    // coalesced loads, non-temporal (176 MB streamed once; don't evict
    // the reusable y lines from the 192 MB L2). Overlaps the async
    // y-tile transfer. jnp.argmax = first max -> strict '>' updates.
    // ---------------------------------------------------------------
    const int w = w0 + ty;
    const int h = h0 + tx * PXT;
    const v4f* xp = reinterpret_cast<const v4f*>(
        x + (size_t)b * CC * WH + (size_t)w * HH + h);

    v4f v = __builtin_nontemporal_load(xp);
    float b0 = v.x, b1 = v.y, b2 = v.z, b3 = v.w;
    int   m0 = 0,  m1 = 0,  m2 = 0,  m3 = 0;
#pragma unroll
    for (int c = 1; c < CC; ++c) {
        v = __builtin_nontemporal_load(xp + (size_t)c * (WH / 4));
        if (v.x > b0) { b0 = v.x; m0 = c; }
        if (v.y > b1) { b1 = v.y; m1 = c; }
        if (v.z > b2) { b2 = v.z; m2 = c; }
        if (v.w > b3) { b3 = v.w; m3 = c; }
    }

    // All of this wave's async LDS writes landed, then barrier so every
    // wave's tile contribution is visible.
    asm volatile("s_wait_asynccnt 0x0" ::: "memory");
    __syncthreads();

    // ---------------------------------------------------------------
    // 5x5 neighborhood mismatch count. 8 ints per row cover the four
    // overlapping windows of this thread's 4 pixels (row base is 16B
    // aligned: 528B row stride, 16B per-thread step -> ds_load_b128x2).
    // ---------------------------------------------------------------
    int c0 = 0, c1 = 0, c2 = 0, c3 = 0;
#pragma unroll
    for (int dr = 0; dr < 5; ++dr) {
        const int* row = &tile[(ty + dr) * TILE_W + tx * PXT];
        const int v0 = row[0], v1 = row[1], v2 = row[2], v3 = row[3];
        const int v4 = row[4], v5 = row[5], v6 = row[6], v7 = row[7];
        c0 += (v0 != m0) + (v1 != m0) + (v2 != m0) + (v3 != m0) + (v4 != m0);
        c1 += (v1 != m1) + (v2 != m1) + (v3 != m1) + (v4 != m1) + (v5 != m1);
        c2 += (v2 != m2) + (v3 != m2) + (v4 != m2) + (v5 != m2) + (v6 != m2);
        c3 += (v3 != m3) + (v4 != m3) + (v5 != m3) + (v6 != m3) + (v7 != m3);
    }

    v4f o;
    o.x = fmaf(PENALTY, (float)c0, 1.0f);
    o.y = fmaf(PENALTY, (float)c1, 1.0f);
    o.z = fmaf(PENALTY, (float)c2, 1.0f);
    o.w = fmaf(PENALTY, (float)c3, 1.0f);
    // Output is written once, never read back: non-temporal store.
    __builtin_nontemporal_store(
        o, reinterpret_cast<v4f*>(out + (size_t)b * WH + (size_t)w * HH + h));
}

extern "C" void kernel_launch(void* const* d_in, const int* in_sizes, int n_in,
                              void* d_out, int out_size, void* d_ws, size_t ws_size,
                              hipStream_t stream) {
    (void)in_sizes; (void)n_in; (void)d_ws; (void)ws_size; (void)out_size;
    const float* x  = (const float*)d_in[0];   // [8, 21, 512, 512] f32
    const int*   yy = (const int*)d_in[1];     // [8, 512, 512] i32
    float*       o  = (float*)d_out;           // [8, 1, 512, 512] f32

    dim3 block(TX, TY, 1);                                  // 256 = 8 wave32
    dim3 grid(HH / (TX * PXT), WW / TY, 8);                 // (4, 64, 8)
    GetWeightMatrix_25812753449051_kernel<<<grid, block, 0, stream>>>(x, yy, o);
}